// CustomGCN_14980845928717
// MI455X (gfx1250) — compile-verified
//
#include <hip/hip_runtime.h>
#include <hip/hip_bf16.h>

typedef __attribute__((ext_vector_type(2))) float v2f;
typedef __attribute__((ext_vector_type(8))) float v8f;

#define D 128
#define NGRAPH 64
#define NCLASS 6
#define ROWS_PER_BLOCK 64   // 4 m-tiles of 16 rows per block

// ---------------- degree / norm ----------------
__global__ void k_init_deg(float* deg, int n) {
    int i = blockIdx.x * blockDim.x + threadIdx.x;
    if (i < n) deg[i] = 1.0f;  // self-loop contributes 1
}

__global__ void k_count_deg(const int* __restrict__ dst, float* deg, int ne) {
    int e = blockIdx.x * blockDim.x + threadIdx.x;
    if (e < ne) atomicAdd(&deg[dst[e]], 1.0f);
}

__global__ void k_finish_dinv(float* deg, int n) {
    int i = blockIdx.x * blockDim.x + threadIdx.x;
    if (i < n) {
        float d = deg[i];
        deg[i] = (d > 0.0f) ? rsqrtf(d) : 0.0f;
    }
}

// ---------------- fp32 WMMA GEMM: C[nrows x 128] = A[nrows x 128] @ W[128 x 128] ----------------
// Block = 8 waves, 64 rows x 128 cols of output.
//  * A tile (64x128 f32, 32 KB) staged in LDS once per block via async global->LDS copy.
//  * Wave w owns cols [16w, 16w+16) and iterates 4 row-tiles, so each B fragment
//    (loaded once from global) feeds 4 WMMAs; A fragments come from LDS (ds_load_b64).
// V_WMMA_F32_16X16X4_F32 per-lane layout (ISA 7.12.2):
//   A (16x4):  lane L -> M = L&15, K = (L>>4)*2 + {0,1}
//   B (4x16):  lane L -> N = L&15, K = (L>>4)*2 + {0,1}
//   C (16x16): VGPR v, lane L -> M = (L>>4)*8 + v, N = L&15
__global__ __launch_bounds__(256) void k_gemm_wmma_f32(
    const float* __restrict__ A, const float* __restrict__ W,
    float* __restrict__ C, int nrows) {
    __shared__ float tile[ROWS_PER_BLOCK * D];   // 32 KB

    const int wave = threadIdx.x >> 5;
    const int lane = threadIdx.x & 31;
    const int row0 = blockIdx.x * ROWS_PER_BLOCK;
    if (row0 >= nrows) return;

    // --- async copy of the A tile into LDS (ASYNCcnt path) ---
    {
        const float* gbase = A + (size_t)row0 * D;
        unsigned ldsBase = (unsigned)(size_t)&tile[0];
        #pragma unroll
        for (int i = 0; i < 8; ++i) {
            int idx4 = threadIdx.x + i * 256;        // float4 index; row = idx4/32
            if (row0 + (idx4 >> 5) < nrows) {
                unsigned ldsOff = ldsBase + (unsigned)idx4 * 16u;
                const float* g = gbase + (size_t)idx4 * 4;
                asm volatile("global_load_async_to_lds_b128 %0, %1, off"
                             :: "v"(ldsOff), "v"(g) : "memory");
            }
        }
        asm volatile("s_wait_asynccnt 0x0" ::: "memory");
    }
    __syncthreads();

    const int col0 = wave * 16;
    const int n    = lane & 15;
    const int koff = (lane >> 4) * 2;

    v8f acc[4] = {{}, {}, {}, {}};
    #pragma unroll 4
    for (int k0 = 0; k0 < D; k0 += 4) {
        const int k = k0 + koff;
        v2f b;
        b.x = W[(size_t)k * D + col0 + n];
        b.y = W[(size_t)(k + 1) * D + col0 + n];
        #pragma unroll
        for (int mt = 0; mt < 4; ++mt) {
            const float* arow = tile + (mt * 16 + n) * D;   // row M = n (lane&15)
            v2f a;
            a.x = arow[k];
            a.y = arow[k + 1];      // consecutive -> ds_load_b64
            acc[mt] = __builtin_amdgcn_wmma_f32_16x16x4_f32(
                false, a, false, b, (short)0, acc[mt], false, false);
        }
    }

    const int mbase = (lane >> 4) * 8;
    #pragma unroll
    for (int mt = 0; mt < 4; ++mt) {
        int rbase = row0 + mt * 16;
        #pragma unroll
        for (int v = 0; v < 8; ++v) {
            int r = rbase + mbase + v;
            if (r < nrows) C[(size_t)r * D + col0 + n] = acc[mt][v];
        }
    }
}

// ---------------- edge scatter: agg[dst] += norm * hw[src] ----------------
// One wave per edge; each lane moves a float4 (32 lanes * 4 = 128 cols).
__global__ void k_scatter(const int* __restrict__ src, const int* __restrict__ dst,
                          const float* __restrict__ hw, const float* __restrict__ dinv,
                          float* __restrict__ agg, int ne) {
    int t    = blockIdx.x * blockDim.x + threadIdx.x;
    int e    = t >> 5;
    int lane = t & 31;
    if (e >= ne) return;
    int s = src[e], d = dst[e];
    float norm = dinv[s] * dinv[d];
    const float4* hv = (const float4*)(hw + (size_t)s * D);
    float4 v = hv[lane];
    float* ad = agg + (size_t)d * D + lane * 4;
    atomicAdd(ad + 0, v.x * norm);
    atomicAdd(ad + 1, v.y * norm);
    atomicAdd(ad + 2, v.z * norm);
    atomicAdd(ad + 3, v.w * norm);
}

// ---------------- fused self-loop + bias + relu ----------------
__global__ void k_bias_relu(const float* __restrict__ agg, const float* __restrict__ hw,
                            const float* __restrict__ dinv, const float* __restrict__ b,
                            float* __restrict__ out, int n) {
    int t = blockIdx.x * blockDim.x + threadIdx.x;
    int node = t >> 7;           // D == 128
    int c    = t & (D - 1);
    if (node >= n) return;
    float di = dinv[node];
    size_t idx = (size_t)node * D + c;
    float v = agg[idx] + hw[idx] * (di * di) + b[c];
    out[idx] = fmaxf(v, 0.0f);
}

// ---------------- global mean pool ----------------
__global__ void k_pool(const float* __restrict__ h, const int* __restrict__ batch,
                       float* __restrict__ sums, float* __restrict__ counts, int n) {
    int t    = blockIdx.x * blockDim.x + threadIdx.x;
    int node = t >> 5;
    int lane = t & 31;
    if (node >= n) return;
    int g = batch[node];
    const float4* hv = (const float4*)(h + (size_t)node * D);
    float4 v = hv[lane];
    float* s = sums + (size_t)g * D + lane * 4;
    atomicAdd(s + 0, v.x);
    atomicAdd(s + 1, v.y);
    atomicAdd(s + 2, v.z);
    atomicAdd(s + 3, v.w);
    if (lane == 0) atomicAdd(&counts[g], 1.0f);
}

__global__ void k_finalize_emb(const float* __restrict__ sums, const float* __restrict__ counts,
                               float* __restrict__ out_emb) {
    int g = blockIdx.x;      // 64 blocks
    int c = threadIdx.x;     // 128 threads
    float cnt = fmaxf(counts[g], 1.0f);
    out_emb[(size_t)g * D + c] = sums[(size_t)g * D + c] / cnt;
}

__global__ void k_logits(const float* __restrict__ emb, const float* __restrict__ linW,
                         const float* __restrict__ linb, float* __restrict__ out_logits) {
    int t = blockIdx.x * blockDim.x + threadIdx.x;
    if (t >= NGRAPH * NCLASS) return;
    int g = t / NCLASS, c = t % NCLASS;
    float acc = linb[c];
    #pragma unroll 4
    for (int k = 0; k < D; ++k)
        acc += emb[(size_t)g * D + k] * linW[(size_t)k * NCLASS + c];
    out_logits[(size_t)g * NCLASS + c] = acc;
}

extern "C" void kernel_launch(void* const* d_in, const int* in_sizes, int n_in,
                              void* d_out, int out_size, void* d_ws, size_t ws_size,
                              hipStream_t stream) {
    const float* x     = (const float*)d_in[0];
    const int*   eidx  = (const int*)d_in[1];
    const int*   batch = (const int*)d_in[2];
    const float* Ws[3] = {(const float*)d_in[3], (const float*)d_in[5], (const float*)d_in[7]};
    const float* bs[3] = {(const float*)d_in[4], (const float*)d_in[6], (const float*)d_in[8]};
    const float* linW  = (const float*)d_in[9];
    const float* linb  = (const float*)d_in[10];

    const int nnodes = in_sizes[2];          // batch has one entry per node
    const int nedges = in_sizes[1] / 2;      // edge_index is [2, E]
    const int* esrc = eidx;
    const int* edst = eidx + nedges;

    float* out_logits = (float*)d_out;                    // [64, 6]
    float* out_emb    = (float*)d_out + NGRAPH * NCLASS;  // [64, 128]

    // workspace carve-up
    float*  ws    = (float*)d_ws;
    const size_t NH = (size_t)nnodes * D;
    float* dinv   = ws;                      // [N]
    float* hw     = dinv + nnodes;           // [N, D]
    float* agg    = hw + NH;                 // [N, D]
    float* hbuf   = agg + NH;                // [N, D]
    float* psum   = hbuf + NH;               // [64, D]
    float* pcount = psum + NGRAPH * D;       // [64]

    const int TB = 256;
    auto cdiv = [](long long a, long long b) { return (int)((a + b - 1) / b); };

    // 1) degree -> dinv
    k_init_deg<<<cdiv(nnodes, TB), TB, 0, stream>>>(dinv, nnodes);
    k_count_deg<<<cdiv(nedges, TB), TB, 0, stream>>>(edst, dinv, nedges);
    k_finish_dinv<<<cdiv(nnodes, TB), TB, 0, stream>>>(dinv, nnodes);

    // 2) three GCN layers
    const float* hin = x;
    const int gemm_blocks = cdiv(nnodes, ROWS_PER_BLOCK);
    for (int l = 0; l < 3; ++l) {
        k_gemm_wmma_f32<<<gemm_blocks, 256, 0, stream>>>(hin, Ws[l], hw, nnodes);
        hipMemsetAsync(agg, 0, NH * sizeof(float), stream);
        k_scatter<<<cdiv((long long)nedges * 32, TB), TB, 0, stream>>>(
            esrc, edst, hw, dinv, agg, nedges);
        k_bias_relu<<<cdiv((long long)nnodes * D, TB), TB, 0, stream>>>(
            agg, hw, dinv, bs[l], hbuf, nnodes);
        hin = hbuf;
    }

    // 3) mean pool + classifier
    hipMemsetAsync(psum, 0, (NGRAPH * D + NGRAPH) * sizeof(float), stream);
    k_pool<<<cdiv((long long)nnodes * 32, TB), TB, 0, stream>>>(hbuf, batch, psum, pcount, nnodes);
    k_finalize_emb<<<NGRAPH, D, 0, stream>>>(psum, pcount, out_emb);
    k_logits<<<cdiv(NGRAPH * NCLASS, TB), TB, 0, stream>>>(out_emb, linW, linb, out_logits);
}